// QuantumClassifier_73967926772387
// MI455X (gfx1250) — compile-verified
//
#include <hip/hip_runtime.h>

typedef __attribute__((ext_vector_type(2))) float v2f;
typedef __attribute__((ext_vector_type(8))) float v8f;

#define NQ 4
#define NLAYERS 2

// ---------------------------------------------------------------------------
// Kernel 1 (runs once, 16 threads): build the fixed 16x16 variational-layer
// matrix U from weights with PRECISE sincos. Thread j simulates the circuit
// on basis vector e_j (full 16-state in registers).
// Bit convention: flattened index k = q0*8 + q1*4 + q2*2 + q3 (q0 = MSB),
// so wire i lives at bit position (3 - i).
// ---------------------------------------------------------------------------
__global__ void __launch_bounds__(32) pqc_build_U(const float* __restrict__ w,
                                                  float* __restrict__ U) {
  const int j = threadIdx.x;
  if (j >= 16) return;

  float st[16];
#pragma unroll
  for (int k = 0; k < 16; ++k) st[k] = (k == j) ? 1.0f : 0.0f;

#pragma unroll
  for (int layer = 0; layer < NLAYERS; ++layer) {
    // RY(weights[layer][i]) on each wire
#pragma unroll
    for (int i = 0; i < NQ; ++i) {
      const float h = 0.5f * w[layer * NQ + i];
      float s, c;
      sincosf(h, &s, &c);  // precise: U is reused by every sample
      const int m = 1 << (3 - i);
#pragma unroll
      for (int k = 0; k < 16; ++k) {
        if ((k & m) == 0) {
          const int k1 = k | m;
          const float a = st[k], b = st[k1];
          st[k]  = c * a - s * b;   // new0 = c*s0 - s*s1
          st[k1] = s * a + c * b;   // new1 = s*s0 + c*s1
        }
      }
    }
    // CNOT ring: (0,1) (1,2) (2,3) (3,0); swap amplitudes where ctrl bit = 1
    const int ctrl[4] = {0, 1, 2, 3};
    const int tgt[4]  = {1, 2, 3, 0};
#pragma unroll
    for (int e = 0; e < 4; ++e) {
      const int mc = 1 << (3 - ctrl[e]);
      const int mt = 1 << (3 - tgt[e]);
#pragma unroll
      for (int k = 0; k < 16; ++k) {
        if ((k & mc) != 0 && (k & mt) == 0) {
          const int k1 = k | mt;
          const float t = st[k];
          st[k] = st[k1];
          st[k1] = t;
        }
      }
    }
  }
  // U column j (row-major U[k][j])
#pragma unroll
  for (int k = 0; k < 16; ++k) U[k * 16 + j] = st[k];
}

// ---------------------------------------------------------------------------
// Kernel 2 (streaming, memory-bound): per 16-sample tile, build the
// product-state operand S (B-matrix) from 4 fast HW sincos (args are
// 0.5*N(0,1): tiny, no range-reduction hazard), compute V = U*S with
// 4x v_wmma_f32_16x16x4_f32, then
// out = (sum_{k<8} V_k^2 - sum_{k>=8} V_k^2 + 1) * 0.5.
// ---------------------------------------------------------------------------
__global__ void __launch_bounds__(256) pqc_wmma(const float* __restrict__ x,
                                                const float* __restrict__ U,
                                                float* __restrict__ out,
                                                int nsamp) {
  const int lane = threadIdx.x & 31;
  const int hi   = lane >> 4;   // lane half: selects K sub-rows / M half
  const int col  = lane & 15;   // sample-in-tile for B/C; row M for A
  const int wave   = blockIdx.x * (blockDim.x >> 5) + (threadIdx.x >> 5);
  const int nwaves = gridDim.x * (blockDim.x >> 5);
  const int ntiles = (nsamp + 15) >> 4;

  // A operand (constant across tiles): lane holds U[col][4c + 2*hi + {0,1}]
  v2f a[4];
#pragma unroll
  for (int c = 0; c < 4; ++c) {
    const int k0 = 4 * c + 2 * hi;
    a[c].x = U[col * 16 + k0];
    a[c].y = U[col * 16 + k0 + 1];
  }

  for (int tile = wave; tile < ntiles; tile += nwaves) {
    const int sidx = tile * 16 + col;
    const int sclamp = (sidx < nsamp) ? sidx : (nsamp - 1);

    // encoding angles -> 4 (cos, sin) pairs via HW trans ops
    const float4 xv = *reinterpret_cast<const float4*>(x + (size_t)sclamp * 4);
    float g[4][2];  // g[i][0]=cos(x_i/2), g[i][1]=sin(x_i/2)
    __sincosf(0.5f * xv.x, &g[0][1], &g[0][0]);
    __sincosf(0.5f * xv.y, &g[1][1], &g[1][0]);
    __sincosf(0.5f * xv.z, &g[2][1], &g[2][0]);
    __sincosf(0.5f * xv.w, &g[3][1], &g[3][0]);

    // Product-state amplitudes. For k = 4c + 2*hi + v:
    //   bit0 = v, bit1 = hi (constant per lane), bits2..3 = c
    // s_k = g0[c>>1] * g1[c&1] * g2[hi] * g3[v]
    const float g2h = g[2][hi];
    v2f b[4];
#pragma unroll
    for (int c = 0; c < 4; ++c) {
      const float t = g[0][(c >> 1) & 1] * g[1][c & 1] * g2h;
      b[c].x = t * g[3][0];
      b[c].y = t * g[3][1];
    }

    // V = U * S  (K = 16 via four 16x16x4 f32 WMMAs)
    v8f acc = {};
#pragma unroll
    for (int c = 0; c < 4; ++c) {
      acc = __builtin_amdgcn_wmma_f32_16x16x4_f32(
          /*neg_a=*/false, a[c], /*neg_b=*/false, b[c],
          /*c_mod=*/(short)0, acc, /*reuse_a=*/false, /*reuse_b=*/false);
    }

    // lanes 0-15 hold M=0..7 (Z0=+1), lanes 16-31 hold M=8..15 (Z0=-1)
    float p = 0.0f;
#pragma unroll
    for (int r = 0; r < 8; ++r) p += acc[r] * acc[r];
    const float q = __shfl_down(p, 16, 32);  // partner half's probability sum
    if (hi == 0 && sidx < nsamp) out[sidx] = (p - q + 1.0f) * 0.5f;
  }
}

// ---------------------------------------------------------------------------
extern "C" void kernel_launch(void* const* d_in, const int* in_sizes, int n_in,
                              void* d_out, int out_size, void* d_ws, size_t ws_size,
                              hipStream_t stream) {
  const float* x = (const float*)d_in[0];       // [B, 4] f32
  const float* w = (const float*)d_in[1];       // [2, 4] f32
  float* out = (float*)d_out;                   // [B] f32
  float* U   = (float*)d_ws;                    // 16*16 f32 scratch (1 KB)
  const int nsamp = in_sizes[0] / NQ;

  pqc_build_U<<<1, 32, 0, stream>>>(w, U);

  const int threads = 256;                      // 8 waves/block
  const int ntiles = (nsamp + 15) >> 4;
  int blocks = 2048;
  const int max_blocks = (ntiles + (threads >> 5) - 1) / (threads >> 5);
  if (blocks > max_blocks) blocks = max_blocks;
  if (blocks < 1) blocks = 1;
  pqc_wmma<<<blocks, threads, 0, stream>>>(x, U, out, nsamp);
}